// GMDRSSMStep_62302795596649
// MI455X (gfx1250) — compile-verified
//
#include <hip/hip_runtime.h>
#include <math.h>

typedef float v2f __attribute__((ext_vector_type(2)));
typedef float v8f __attribute__((ext_vector_type(8)));
typedef unsigned int u32x4 __attribute__((ext_vector_type(4)));
typedef int i32x4 __attribute__((ext_vector_type(4)));
typedef int i32x8 __attribute__((ext_vector_type(8)));

#define NDIM 4
#define D_IN 128
#define KS 3
#define RANK 2
#define NPROJ 15              // 3 + 8 + 4 rows, padded to 16
#define ROWS_PER_BLOCK 64     // 2 waves * 32 rows
#define THREADS 64
#define PSCALE 0.1f

#if defined(__has_builtin)
#if __has_builtin(__builtin_amdgcn_tensor_load_to_lds) && \
    __has_builtin(__builtin_amdgcn_s_wait_tensorcnt)
#define HAVE_TDM 1
#endif
#endif

__global__ __launch_bounds__(THREADS)
void gmdr_ssm_step_kernel(const float* __restrict__ h,
                          const float* __restrict__ x,
                          const float* __restrict__ W_alpha,
                          const float* __restrict__ b_alpha,
                          const float* __restrict__ W_a,
                          const float* __restrict__ b_a,
                          const float* __restrict__ W_B,
                          const float* __restrict__ b_B,
                          float* __restrict__ out,
                          int batch)
{
    __shared__ __attribute__((aligned(16))) float Wlds[16 * D_IN];              // combined weights [16][128]
    __shared__ __attribute__((aligned(16))) float xlds[ROWS_PER_BLOCK * D_IN];  // x tile [64][128]
    __shared__ __attribute__((aligned(16))) float projlds[ROWS_PER_BLOCK * 16]; // projections [64][16]

    const int tid  = threadIdx.x;
    const int wave = tid >> 5;
    const int lane = tid & 31;
    const int blockRow = blockIdx.x * ROWS_PER_BLOCK;

    // ---- stage x tile: global -> LDS ----
#ifdef HAVE_TDM
    // One Tensor-Data-Mover DMA per block: 64x128 f32 tile, row-major,
    // straight into LDS (no VGPR transit). Tail rows read OOB -> zero.
    if (wave == 0) {
        const unsigned long long gaddr =
            (unsigned long long)(const void*)(x + (long)blockRow * D_IN);
        const unsigned ldsaddr = (unsigned)(unsigned long long)
            (__attribute__((address_space(3))) float*)xlds;
        const int remain = batch - blockRow;
        const unsigned rows =
            (unsigned)(remain < ROWS_PER_BLOCK ? remain : ROWS_PER_BLOCK);

        u32x4 g0;
        g0.x = 0x1u;                                   // count=1 (valid user D#)
        g0.y = ldsaddr;                                // lds_addr [63:32]
        g0.z = (unsigned)(gaddr & 0xffffffffu);        // global_addr [95:64]
        g0.w = (unsigned)((gaddr >> 32) & 0x01ffffffu) // global_addr[56:32]
             | 0x80000000u;                            // type=2 @ bits[127:126]

        i32x8 g1;
        g1[0] = 0x20000;                 // wg_mask=0, data_size=2 (4 bytes)
        g1[1] = (int)(128u << 16);       // tensor_dim0[15:0]=128 @ bits[63:48]
        g1[2] = (int)(rows << 16);       // dim0 hi=0; tensor_dim1[15:0]=rows
        g1[3] = (int)(128u << 16);       // dim1 hi=0; tile_dim0=128
        g1[4] = ROWS_PER_BLOCK;          // tile_dim1=64, tile_dim2=0
        g1[5] = D_IN;                    // tensor_dim0_stride=128
        g1[6] = 0;                       // stride hi / dim1_stride (2D: unused)
        g1[7] = 0;

        const i32x4 z4 = {0, 0, 0, 0};
        const i32x8 z8 = {0, 0, 0, 0, 0, 0, 0, 0};
        __builtin_amdgcn_tensor_load_to_lds(g0, g1, z4, z4, z8, 0);
    }
#else
    {
        const float4* xg = (const float4*)x;
        float4* xl4 = (float4*)xlds;
        const long base4  = (long)blockRow * (D_IN / 4);
        const long limit4 = (long)batch * (D_IN / 4);
        for (int i = tid; i < ROWS_PER_BLOCK * D_IN / 4; i += THREADS) {
            const long g = base4 + i;
            float4 v = make_float4(0.f, 0.f, 0.f, 0.f);
            if (g < limit4) v = xg[g];
            xl4[i] = v;
        }
    }
#endif

    // ---- stage combined weight matrix Wc[16][128] into LDS (overlaps TDM) ----
    for (int i = tid; i < 16 * D_IN; i += THREADS) {
        const int r = i >> 7;          // row 0..15
        const int c = i & (D_IN - 1);  // col 0..127
        float v;
        if (r < KS)                    v = W_alpha[r * D_IN + c];
        else if (r < KS + RANK * NDIM) v = W_a[(r - KS) * D_IN + c];
        else if (r < NPROJ)            v = W_B[(r - KS - RANK * NDIM) * D_IN + c];
        else                           v = 0.0f;
        Wlds[i] = v;
    }

#ifdef HAVE_TDM
    if (wave == 0) {
        __builtin_amdgcn_s_wait_tensorcnt(0);  // issuing wave waits for the DMA
    }
#endif
    __syncthreads();                           // publish LDS tile to all waves

    // ---- WMMA: proj(16x16) = Wc(16x128) * x_tile^T(128x16), two tiles/wave ----
    // A frag (16x4 f32): lane l -> M = l&15, VGPR v holds K = 4c + 2*(l>>4) + v
    // B frag (4x16 f32): lane l -> N = l&15, VGPR v holds K = 4c + 2*(l>>4) + v
    const int m       = lane & 15;
    const int khalf   = (lane >> 4) * 2;
    const int waveRow = wave * 32;

    v8f acc0 = {0.f, 0.f, 0.f, 0.f, 0.f, 0.f, 0.f, 0.f};
    v8f acc1 = {0.f, 0.f, 0.f, 0.f, 0.f, 0.f, 0.f, 0.f};

    for (int c = 0; c < D_IN / 4; ++c) {
        const int koff = c * 4 + khalf;
        v2f aF = *(const v2f*)&Wlds[m * D_IN + koff];
        v2f b0 = *(const v2f*)&xlds[(waveRow + m) * D_IN + koff];
        v2f b1 = *(const v2f*)&xlds[(waveRow + 16 + m) * D_IN + koff];
        acc0 = __builtin_amdgcn_wmma_f32_16x16x4_f32(false, aF, false, b0,
                                                     (short)0, acc0, false, false);
        acc1 = __builtin_amdgcn_wmma_f32_16x16x4_f32(false, aF, false, b1,
                                                     (short)0, acc1, false, false);
    }

    // ---- scatter D tiles so each lane owns one row's projections ----
    // D layout: lane l -> column N = l&15, VGPR v -> M = v + 8*(l>>4)
    {
        const int mbase = (lane >> 4) * 8;
        const int r0 = waveRow + m;
        const int r1 = waveRow + 16 + m;
#pragma unroll
        for (int v = 0; v < 8; ++v) {
            projlds[r0 * 16 + mbase + v] = acc0[v];
            projlds[r1 * 16 + mbase + v] = acc1[v];
        }
    }
    __syncthreads();

    // ---- epilogue: one lane per batch row ----
    const int  localRow = waveRow + lane;
    const long row      = (long)blockRow + localRow;
    if (row < batch) {
        float p[NPROJ];
#pragma unroll
        for (int i = 0; i < NPROJ; ++i) p[i] = projlds[localRow * 16 + i];

        // softmax over 3 kernel-neighborhood logits
        const float l0 = p[0] + b_alpha[0];
        const float l1 = p[1] + b_alpha[1];
        const float l2 = p[2] + b_alpha[2];
        const float mx = fmaxf(l0, fmaxf(l1, l2));
        const float e0 = expf(l0 - mx), e1 = expf(l1 - mx), e2 = expf(l2 - mx);
        const float inv = 1.0f / (e0 + e1 + e2);
        const float al0 = e0 * inv, al1 = e1 * inv, al2 = e2 * inv;
        const float als = al0 + al1 + al2;

        // low-rank tanh perturbation coefficients a[r][i]
        float a0[NDIM], a1[NDIM];
#pragma unroll
        for (int i = 0; i < NDIM; ++i) {
            a0[i] = tanhf(p[3 + i] + b_a[i])        * PSCALE;
            a1[i] = tanhf(p[7 + i] + b_a[NDIM + i]) * PSCALE;
        }

        // Bx + b_B
        const float bx0 = p[11] + b_B[0];
        const float bx1 = p[12] + b_B[1];
        const float bx2 = p[13] + b_B[2];
        const float bx3 = p[14] + b_B[3];

        const float4 hv = ((const float4*)h)[row];

        // A = al0*I + al1*Swap01 + al2*SignFlip0 + a0.*C3 + a1.*C4 (closed form)
        float4 o;
        o.x = (al0 - al2) * hv.x + (al1 + a0[0] + a1[0]) * hv.y + bx0;
        o.y = al1 * hv.x + (al0 + al2) * hv.y + (a0[1] + a1[1]) * hv.z + bx1;
        o.z = a0[2] * hv.x + als * hv.z + a1[2] * hv.w + bx2;
        o.w = a1[3] * hv.x + (als + a0[3]) * hv.w + bx3;

        ((float4*)out)[row] = o;
    }
}

extern "C" void kernel_launch(void* const* d_in, const int* in_sizes, int n_in,
                              void* d_out, int out_size, void* d_ws, size_t ws_size,
                              hipStream_t stream) {
    const float* h       = (const float*)d_in[0];
    const float* x       = (const float*)d_in[1];
    const float* W_alpha = (const float*)d_in[2];
    const float* b_alpha = (const float*)d_in[3];
    const float* W_a     = (const float*)d_in[4];
    const float* b_a     = (const float*)d_in[5];
    const float* W_B     = (const float*)d_in[6];
    const float* b_B     = (const float*)d_in[7];
    float* out = (float*)d_out;

    const int batch = in_sizes[0] / NDIM;   // h is (BATCH, 4)
    const int grid  = (batch + ROWS_PER_BLOCK - 1) / ROWS_PER_BLOCK;

    gmdr_ssm_step_kernel<<<grid, THREADS, 0, stream>>>(
        h, x, W_alpha, b_alpha, W_a, b_a, W_B, b_B, out, batch);
}